// MultiScaleRetention_48189533061454
// MI455X (gfx1250) — compile-verified
//
#include <hip/hip_runtime.h>
#include <hip/hip_bf16.h>
#include <math.h>

typedef __attribute__((ext_vector_type(2))) float v2f;
typedef __attribute__((ext_vector_type(8))) float v8f;
typedef __attribute__((ext_vector_type(4))) unsigned int u32x4;
typedef __attribute__((ext_vector_type(4))) int i32x4;
typedef __attribute__((ext_vector_type(8))) int i32x8;

#define D_MODEL 1024
#define N_HEADS 16
#define D_HEAD  64
#define CHUNK_C 128
#define BATCH   4
#define SEQLEN  4096
#define NKEYPAD 144   // 129 keys padded to 9 WMMA n-tiles
#define APITCH  145   // odd-DWORD pitch -> conflict-free LDS A-frag reads

#define KSTEP   32    // K tile depth staged in LDS per TDM transfer
#define XPITCH  33    // 32 + 1 pad DWORD (TDM pad_interval=32, pad_amount=1)
#define WPITCH  65    // 64 + 1 pad DWORD (TDM pad_interval=64, pad_amount=1)

static __device__ __forceinline__
v8f wmma_f32x4(v2f a, v2f b, v8f c) {
  // D(16x16,f32) = A(16x4,f32) x B(4x16,f32) + C
  return __builtin_amdgcn_wmma_f32_16x16x4_f32(false, a, false, b,
                                               (short)0, c, false, false);
}

// ---------------------------------------------------------------------------
// Issue one TDM 2D tile load: global (row-major, row_stride elems of 4B)
// -> LDS at lds_byte_off, tile_d0 contiguous elems per row, tile_d1 rows.
// pad_code: pad_interval code (4 => every 32 DWORDs, 5 => every 64 DWORDs);
// pad_amount code 0 => insert 1 DWORD, giving odd LDS pitch.
// D# packing per CDNA5 ISA.
// Toolchain note: clang-23 lane -> 6-arg builtin (extra int32x8 group).
// ---------------------------------------------------------------------------
static __device__ __forceinline__
void tdm_load_2d(const float* gptr, unsigned lds_byte_off,
                 unsigned tile_d0, unsigned tile_d1,
                 unsigned row_stride, unsigned pad_code) {
  const unsigned long long ga = (unsigned long long)(size_t)gptr;

  u32x4 g0;
  g0[0] = 1u;                                           // count=1, user D#
  g0[1] = lds_byte_off;                                 // lds_addr
  g0[2] = (unsigned)(ga & 0xFFFFFFFFu);                 // global_addr[31:0]
  g0[3] = (unsigned)((ga >> 32) & 0x01FFFFFFu)          // global_addr[56:32]
        | (2u << 30);                                   // type = 2 ("image")

  const unsigned td0 = row_stride;                      // tensor_dim0 (elems)
  const unsigned td1 = 0x00100000u;                     // tensor_dim1 (large)
  i32x8 g1;
  g1[0] = (int)((2u << 16)                              // data_size: 4 bytes
              | (1u << 20)                              // pad_enable
              | (pad_code << 22)                        // pad_interval
              | (0u << 25));                            // pad_amount: 1 DWORD
  g1[1] = (int)((td0 & 0xFFFFu) << 16);                 // tensor_dim0[15:0]
  g1[2] = (int)((td0 >> 16) | ((td1 & 0xFFFFu) << 16)); // td0 hi | td1 lo
  g1[3] = (int)((td1 >> 16) | (tile_d0 << 16));         // td1 hi | tile_dim0
  g1[4] = (int)tile_d1;                                 // tile_dim1 (tile_dim2=0)
  g1[5] = (int)row_stride;                              // tensor_dim0_stride lo32
  g1[6] = 0;                                            // stride hi, stride1 lo
  g1[7] = 0;

  const i32x4 z4 = {0, 0, 0, 0};                        // 2D: groups 2/3 unused
  const i32x8 z8 = {0, 0, 0, 0, 0, 0, 0, 0};
  __builtin_amdgcn_tensor_load_to_lds(g0, g1, z4, z4, z8, 0);
}

static __device__ __forceinline__ unsigned lds_off(const void* p) {
  // LDS aperture: generic addr low 32 bits == LDS byte offset
  return (unsigned)(size_t)p;
}

// ---------------------------------------------------------------------------
// Y[M,N] = X[M,K] @ W[K,N] + bias[N]
// Block = 8 waves: 128(M) x 64(N) tile. K staged in LDS via TDM, double-
// buffered; wave 0 drives the DMA, everyone syncs on barrier after
// s_wait_tensorcnt. WMMA loop touches only LDS (conflict-free odd pitches).
// ---------------------------------------------------------------------------
__global__ __launch_bounds__(256)
void gemm_bias_f32(const float* __restrict__ X, const float* __restrict__ W,
                   const float* __restrict__ bias, float* __restrict__ Y,
                   int M, int N, int K) {
  __shared__ float Xs[2][128 * XPITCH];   // 2 x 16.5 KB
  __shared__ float Ws[2][KSTEP * WPITCH]; // 2 x  8.1 KB

  const int lane   = threadIdx.x & 31;
  const int wave   = threadIdx.x >> 5;
  const int nbase  = blockIdx.x * 64;
  const int mblock = blockIdx.y * 128;
  const int nlane  = lane & 15;         // N index (B/C/D) or M index (A)
  const int khalf  = (lane >> 4) << 1;  // per-half-wave K sub-offset
  const int mhalf  = (lane >> 4) * 8;   // C/D row offset for upper half-wave
  const int nK     = K / KSTEP;

  v8f c[4] = {};

  if (threadIdx.x < 32) {               // wave 0 drives the TDM
    tdm_load_2d(X + (size_t)mblock * K, lds_off(&Xs[0][0]),
                KSTEP, 128, (unsigned)K, 4);
    tdm_load_2d(W + nbase, lds_off(&Ws[0][0]),
                64, KSTEP, (unsigned)N, 5);
  }

  for (int kt = 0; kt < nK; ++kt) {
    const int cur = kt & 1;
    if (threadIdx.x < 32) {
      if (kt + 1 < nK) {
        const int nxt = cur ^ 1;
        tdm_load_2d(X + (size_t)mblock * K + (kt + 1) * KSTEP,
                    lds_off(&Xs[nxt][0]), KSTEP, 128, (unsigned)K, 4);
        tdm_load_2d(W + (size_t)(kt + 1) * KSTEP * N + nbase,
                    lds_off(&Ws[nxt][0]), 64, KSTEP, (unsigned)N, 5);
        __builtin_amdgcn_s_wait_tensorcnt(2);  // in-order: current tile done
      } else {
        __builtin_amdgcn_s_wait_tensorcnt(0);
      }
    }
    __syncthreads();                    // current LDS tiles visible to all

    const float* xrow = &Xs[cur][(wave * 16 + nlane) * XPITCH];
    #pragma unroll
    for (int kk = 0; kk < KSTEP; kk += 4) {
      v2f a;
      a.x = xrow[kk + khalf];
      a.y = xrow[kk + khalf + 1];
      #pragma unroll
      for (int nt = 0; nt < 4; ++nt) {
        const float* wp = &Ws[cur][(kk + khalf) * WPITCH + nt * 16 + nlane];
        v2f bf;
        bf.x = wp[0];
        bf.y = wp[WPITCH];
        c[nt] = wmma_f32x4(a, bf, c[nt]);
      }
    }
    __syncthreads();                    // done reading before TDM rewrites
  }

  #pragma unroll
  for (int nt = 0; nt < 4; ++nt) {
    const int col = nbase + nt * 16 + nlane;
    const float bb = bias[col];
    #pragma unroll
    for (int r = 0; r < 8; ++r) {
      const int row = mblock + wave * 16 + r + mhalf;
      Y[(size_t)row * N + col] = c[nt][r] + bb;
    }
  }
}

// ---------------------------------------------------------------------------
// Chunked retention attention. One wave handles (b, h, chunk, 16-row q block).
// kf/vf = [carry token at key 0, then 128 chunk tokens], padded to 144 keys.
// ---------------------------------------------------------------------------
__global__ __launch_bounds__(32)
void retention_attn(const float* __restrict__ qb, const float* __restrict__ kb,
                    const float* __restrict__ vb, const float* __restrict__ decay,
                    float* __restrict__ ao) {
  __shared__ float attnLDS[16 * APITCH];

  const int lane  = threadIdx.x;
  const int w     = blockIdx.x;             // q row block within chunk (0..7)
  const int chunk = blockIdx.y;             // 0..31
  const int b     = blockIdx.z / N_HEADS;
  const int h     = blockIdx.z % N_HEADS;

  const int nlane = lane & 15;
  const int khalf = (lane >> 4) << 1;
  const int mhalf = (lane >> 4) * 8;

  // ---- stage 1: scores[16 x 144] = q_strip @ kf^T -------------------------
  const float* kp[9];
  float kmask[9];
  #pragma unroll
  for (int nt = 0; nt < 9; ++nt) {
    const int key = nt * 16 + nlane;            // key 0 = carry token
    const int g   = chunk * CHUNK_C + key - 1;  // global seq position of key
    const bool valid = (key <= CHUNK_C) && (g >= 0);
    kp[nt]    = kb + ((size_t)b * SEQLEN + (valid ? g : 0)) * D_MODEL + h * D_HEAD;
    kmask[nt] = valid ? 1.0f : 0.0f;            // chunk0/key0 -> k=0 (zero carry)
  }
  const int qrow = chunk * CHUNK_C + w * 16 + nlane;
  const float* qp = qb + ((size_t)b * SEQLEN + qrow) * D_MODEL + h * D_HEAD;

  v8f s[9] = {};
  for (int d0 = 0; d0 < D_HEAD; d0 += 4) {
    v2f a;
    a.x = qp[d0 + khalf];
    a.y = qp[d0 + khalf + 1];
    #pragma unroll
    for (int nt = 0; nt < 9; ++nt) {
      v2f bf;
      bf.x = kp[nt][d0 + khalf]     * kmask[nt];
      bf.y = kp[nt][d0 + khalf + 1] * kmask[nt];
      s[nt] = wmma_f32x4(a, bf, s[nt]);
    }
  }

  // ---- scale * decay^pos, mask padded keys --------------------------------
  const float ldec  = __logf(decay[h]);
  const float scale = 0.125f;                    // D_HEAD^-0.5
  #pragma unroll
  for (int r = 0; r < 8; ++r) {
    const int pos   = w * 16 + r + mhalf;        // q position within chunk
    const float fac = scale * __expf(ldec * (float)pos);
    #pragma unroll
    for (int nt = 0; nt < 9; ++nt) {
      const int key = nt * 16 + nlane;
      s[nt][r] = (key <= CHUNK_C) ? s[nt][r] * fac : -3.0e38f;
    }
  }

  // ---- row softmax: each 16-lane half owns one row per accumulator --------
  #pragma unroll
  for (int r = 0; r < 8; ++r) {
    float mx = s[0][r];
    #pragma unroll
    for (int nt = 1; nt < 9; ++nt) mx = fmaxf(mx, s[nt][r]);
    #pragma unroll
    for (int off = 1; off < 16; off <<= 1) mx = fmaxf(mx, __shfl_xor(mx, off, 32));
    float sum = 0.0f;
    #pragma unroll
    for (int nt = 0; nt < 9; ++nt) {
      const float p = __expf(s[nt][r] - mx);
      s[nt][r] = p;
      sum += p;
    }
    #pragma unroll
    for (int off = 1; off < 16; off <<= 1) sum += __shfl_xor(sum, off, 32);
    const float inv = 1.0f / sum;
    #pragma unroll
    for (int nt = 0; nt < 9; ++nt) s[nt][r] *= inv;
  }

  // ---- spill attn (C/D layout) to LDS, re-read in A-fragment layout -------
  #pragma unroll
  for (int r = 0; r < 8; ++r) {
    const int row = r + mhalf;
    #pragma unroll
    for (int nt = 0; nt < 9; ++nt)
      attnLDS[row * APITCH + nt * 16 + nlane] = s[nt][r];
  }
  __syncthreads();

  // ---- stage 2: out[16 x 64] = attn[16 x 144] @ vf[144 x 64] --------------
  v8f o[4] = {};
  for (int key0 = 0; key0 < NKEYPAD; key0 += 4) {
    const int keyA = key0 + khalf;
    v2f a;
    a.x = attnLDS[nlane * APITCH + keyA];
    a.y = attnLDS[nlane * APITCH + keyA + 1];

    const int g0 = chunk * CHUNK_C + keyA - 1;
    const int g1 = g0 + 1;
    const float m0 = ((keyA     <= CHUNK_C) && (g0 >= 0)) ? 1.0f : 0.0f;
    const float m1 = ((keyA + 1 <= CHUNK_C) && (g1 >= 0)) ? 1.0f : 0.0f;
    const int gc0 = min(max(g0, 0), SEQLEN - 1);
    const int gc1 = min(max(g1, 0), SEQLEN - 1);
    const float* vp0 = vb + ((size_t)b * SEQLEN + gc0) * D_MODEL + h * D_HEAD;
    const float* vp1 = vb + ((size_t)b * SEQLEN + gc1) * D_MODEL + h * D_HEAD;

    #pragma unroll
    for (int nt = 0; nt < 4; ++nt) {
      const int d = nt * 16 + nlane;
      v2f bf;
      bf.x = vp0[d] * m0;    // masked keys have attn == 0 anyway
      bf.y = vp1[d] * m1;
      o[nt] = wmma_f32x4(a, bf, o[nt]);
    }
  }

  // ---- write per-head output strip ---------------------------------------
  #pragma unroll
  for (int nt = 0; nt < 4; ++nt) {
    const int col = h * D_HEAD + nt * 16 + nlane;
    #pragma unroll
    for (int r = 0; r < 8; ++r) {
      const int row = chunk * CHUNK_C + w * 16 + r + mhalf;
      ao[((size_t)b * SEQLEN + row) * D_MODEL + col] = o[nt][r];
    }
  }
}

// ---------------------------------------------------------------------------
extern "C" void kernel_launch(void* const* d_in, const int* in_sizes, int n_in,
                              void* d_out, int out_size, void* d_ws, size_t ws_size,
                              hipStream_t stream) {
  (void)in_sizes; (void)n_in; (void)out_size; (void)ws_size;

  const float* x     = (const float*)d_in[0];
  const float* Wq    = (const float*)d_in[1];
  const float* bq    = (const float*)d_in[2];
  const float* Wk    = (const float*)d_in[3];
  const float* bk    = (const float*)d_in[4];
  const float* Wv    = (const float*)d_in[5];
  const float* bv    = (const float*)d_in[6];
  const float* Wo    = (const float*)d_in[7];
  const float* bo    = (const float*)d_in[8];
  const float* decay = (const float*)d_in[9];

  const int M = BATCH * SEQLEN;              // 16384
  const int D = D_MODEL;                     // 1024
  const size_t plane = (size_t)M * D;        // 16M floats = 64 MB

  float* q  = (float*)d_ws;                  // ws layout: q | k | v | attn_out
  float* k  = q + plane;
  float* v  = k + plane;
  float* ao = v + plane;

  dim3 gG(D / 64, M / 128);                  // (16, 128)
  dim3 bG(256);
  gemm_bias_f32<<<gG, bG, 0, stream>>>(x, Wq, bq, q, M, D, D);
  gemm_bias_f32<<<gG, bG, 0, stream>>>(x, Wk, bk, k, M, D, D);
  gemm_bias_f32<<<gG, bG, 0, stream>>>(x, Wv, bv, v, M, D, D);

  dim3 gA(CHUNK_C / 16, SEQLEN / CHUNK_C, BATCH * N_HEADS);  // (8, 32, 64)
  retention_attn<<<gA, dim3(32), 0, stream>>>(q, k, v, decay, ao);

  gemm_bias_f32<<<gG, bG, 0, stream>>>(ao, Wo, bo, (float*)d_out, M, D, D);
}